// SSD_Loss_25185688224543
// MI455X (gfx1250) — compile-verified
//
#include <hip/hip_runtime.h>
#include <math.h>

#define B_   128
#define P_   8732
#define N_   16
#define C_   81
#define IMG  300.0f
#define THRESH 0.5f

typedef __attribute__((ext_vector_type(2))) float v2f;
typedef __attribute__((ext_vector_type(8))) float v8f;

// ---------------- init: zero accumulators ----------------
__global__ void init_kernel(float* __restrict__ acc, int* __restrict__ npos) {
    if (threadIdx.x < 4) acc[threadIdx.x] = 0.0f;
    if (threadIdx.x == 0) *npos = 0;
}

// ---------------- stage A: matching + SmoothL1 ----------------
// one block per image
__global__ __launch_bounds__(256)
void match_kernel(const float* __restrict__ pred_boxes,   // [B,P,4]
                  const float* __restrict__ tgt_boxes,    // [B,N,4] pixel cxcywh
                  const int*   __restrict__ tgt_labels,   // [B,N]
                  const float* __restrict__ def_boxes,    // [P,4] ratio cxcywh
                  int*   __restrict__ labels_ws,          // [B,P]
                  int*   __restrict__ n_pos_img,          // [B]
                  float* __restrict__ g_accum,            // [0]=sl1 [1]=cls_pos [2]=hardneg
                  int*   __restrict__ g_npos) {
    __shared__ float tcx[N_], tcy[N_], tw[N_], th[N_];
    __shared__ float tx0[N_], ty0[N_], tx1[N_], ty1[N_], tarea[N_];
    __shared__ int   tlab[N_];
    __shared__ unsigned long long tbest[N_];
    __shared__ int   tbidx[N_];
    __shared__ float red_f[256];
    __shared__ int   red_i[256];

    const int b = blockIdx.x, tid = threadIdx.x;
    if (tid < N_) {
        const float* t = tgt_boxes + ((size_t)b * N_ + tid) * 4;
        float cx = t[0] * (1.0f/IMG), cy = t[1] * (1.0f/IMG);
        float w  = t[2] * (1.0f/IMG), h  = t[3] * (1.0f/IMG);
        tcx[tid]=cx; tcy[tid]=cy; tw[tid]=w; th[tid]=h;
        tx0[tid]=cx-0.5f*w; ty0[tid]=cy-0.5f*h;
        tx1[tid]=cx+0.5f*w; ty1[tid]=cy+0.5f*h;
        tarea[tid]=w*h;
        tlab[tid]=tgt_labels[b*N_+tid];
        tbest[tid]=0ull;
    }
    __syncthreads();

    // pass 1: each target's best prior (packed float-bits|index, IoU>=0 so uint order == float order)
    for (int p = tid; p < P_; p += 256) {
        const float* d = def_boxes + (size_t)p * 4;
        float dw=d[2], dh=d[3];
        float dx0=d[0]-0.5f*dw, dy0=d[1]-0.5f*dh, dx1=d[0]+0.5f*dw, dy1=d[1]+0.5f*dh;
        float darea = dw*dh;
        #pragma unroll
        for (int n = 0; n < N_; ++n) {
            float ix0=fmaxf(tx0[n],dx0), iy0=fmaxf(ty0[n],dy0);
            float ix1=fminf(tx1[n],dx1), iy1=fminf(ty1[n],dy1);
            float iw=fmaxf(ix1-ix0,0.f), ih=fmaxf(iy1-iy0,0.f);
            float inter=iw*ih;
            float iou = inter / (tarea[n] + darea - inter);
            unsigned long long pk =
                (((unsigned long long)__float_as_uint(iou)) << 32) | (unsigned int)p;
            atomicMax(&tbest[n], pk);
        }
    }
    __syncthreads();
    if (tid < N_) tbidx[tid] = (int)(tbest[tid] & 0xffffffffull);
    __syncthreads();

    // pass 2: per-prior argmax, scatter override, label + SmoothL1 on positives
    float sl1_loc = 0.f; int pos_loc = 0;
    for (int p = tid; p < P_; p += 256) {
        const float* d = def_boxes + (size_t)p * 4;
        float dcx=d[0], dcy=d[1], dw=d[2], dh=d[3];
        float dx0=dcx-0.5f*dw, dy0=dcy-0.5f*dh, dx1=dcx+0.5f*dw, dy1=dcy+0.5f*dh;
        float darea=dw*dh;
        float best=0.f; int bn=0;
        #pragma unroll
        for (int n = 0; n < N_; ++n) {
            float ix0=fmaxf(tx0[n],dx0), iy0=fmaxf(ty0[n],dy0);
            float ix1=fminf(tx1[n],dx1), iy1=fminf(ty1[n],dy1);
            float iw=fmaxf(ix1-ix0,0.f), ih=fmaxf(iy1-iy0,0.f);
            float inter=iw*ih;
            float iou = inter / (tarea[n] + darea - inter);
            if (iou > best) { best = iou; bn = n; }   // strict: first occurrence like jnp.argmax
        }
        int obj = bn;
        #pragma unroll
        for (int n = 0; n < N_; ++n) if (tbidx[n] == p) obj = n;  // last n wins (scatter semantics)
        int lab = (best > THRESH) ? tlab[obj] : 0;
        labels_ws[(size_t)b * P_ + p] = lab;
        if (lab != 0) {
            ++pos_loc;
            float gx = (tcx[obj]-dcx) * 10.f / dw;
            float gy = (tcy[obj]-dcy) * 10.f / dh;
            float gw = __logf(tw[obj]/dw) * 5.f;
            float gh = __logf(th[obj]/dh) * 5.f;
            const float* pb = pred_boxes + ((size_t)b * P_ + p) * 4;
            float a0=fabsf(pb[0]-gx), a1=fabsf(pb[1]-gy), a2=fabsf(pb[2]-gw), a3=fabsf(pb[3]-gh);
            sl1_loc += (a0<1.f ? 0.5f*a0*a0 : a0-0.5f);
            sl1_loc += (a1<1.f ? 0.5f*a1*a1 : a1-0.5f);
            sl1_loc += (a2<1.f ? 0.5f*a2*a2 : a2-0.5f);
            sl1_loc += (a3<1.f ? 0.5f*a3*a3 : a3-0.5f);
        }
    }
    red_f[tid]=sl1_loc; red_i[tid]=pos_loc; __syncthreads();
    for (int off=128; off>0; off>>=1) {
        if (tid < off) { red_f[tid]+=red_f[tid+off]; red_i[tid]+=red_i[tid+off]; }
        __syncthreads();
    }
    if (tid == 0) {
        n_pos_img[b] = red_i[0];
        atomicAdd(g_npos, red_i[0]);
        atomicAdd(&g_accum[0], red_f[0]);
    }
}

// ---------------- stage B: cross entropy, WMMA sum-of-exp ----------------
// 1 wave = 16 rows; lane (m,sub): classes {4k+2*sub, 4k+2*sub+1} of row m == native
// V_WMMA_F32_16X16X4_F32 A-matrix layout. D = A x ones + C -> row sums of exp.
__global__ __launch_bounds__(256)
void ce_kernel(const float* __restrict__ scores,     // [B,P,C]
               const int*   __restrict__ labels_ws,  // [B*P]
               float* __restrict__ neg_ws,           // [B*P]
               float* __restrict__ g_accum) {
    const int lane = threadIdx.x & 31;
    const int wid  = threadIdx.x >> 5;
    const int m    = lane & 15;
    const int sub  = lane >> 4;
    const long tile = (long)blockIdx.x * 8 + wid;
    const long row  = tile * 16 + m;                 // B_*P_ divisible by 16: no tail
    const float* rp = scores + (size_t)row * C_;

    __builtin_prefetch(scores + ((size_t)row + 16) * C_, 0, 1);

    // load 21 chunks of 2 classes per lane (pad K=81..83 with -inf)
    float x0[21], x1[21];
    #pragma unroll
    for (int k = 0; k < 21; ++k) {
        int c = 4*k + 2*sub;
        x0[k] = (c     < C_) ? rp[c]   : -INFINITY;
        x1[k] = (c + 1 < C_) ? rp[c+1] : -INFINITY;
    }

    // row max (row split across lanes m and m+16)
    float mx = -INFINITY;
    #pragma unroll
    for (int k = 0; k < 21; ++k) { mx = fmaxf(mx, x0[k]); mx = fmaxf(mx, x1[k]); }
    mx = fmaxf(mx, __shfl_xor(mx, 16));

    // score at the target label
    const int lab = labels_ws[row];
    float slab = -INFINITY;
    #pragma unroll
    for (int k = 0; k < 21; ++k) {
        int c = 4*k + 2*sub;
        if (c     == lab) slab = x0[k];
        if (c + 1 == lab) slab = x1[k];
    }
    slab = fmaxf(slab, __shfl_xor(slab, 16));

    // WMMA sum of exp: accumulate 21 chunks of K=4 into 16x16 f32 D
    v8f acc = {};
    v2f ones; ones.x = 1.0f; ones.y = 1.0f;
    #pragma unroll
    for (int k = 0; k < 21; ++k) {
        v2f a;
        a.x = __expf(x0[k] - mx);                    // exp(-inf)=0 handles padding
        a.y = __expf(x1[k] - mx);
        acc = __builtin_amdgcn_wmma_f32_16x16x4_f32(
            false, a, false, ones, (short)0, acc, false, false);
    }

    // D layout: lanes 0-15 vgpr r -> rowsum[r]; lanes 16-31 vgpr r -> rowsum[8+r]
    int r = m & 7;
    float t = acc[0];
    if (r == 1) t = acc[1];
    if (r == 2) t = acc[2];
    if (r == 3) t = acc[3];
    if (r == 4) t = acc[4];
    if (r == 5) t = acc[5];
    if (r == 6) t = acc[6];
    if (r == 7) t = acc[7];
    float sumexp = __shfl(t, r + ((m & 8) << 1));    // src lane holds rowsum[m]

    float ce = (mx + __logf(sumexp)) - slab;         // -logp[label]
    if (sub == 0) {
        bool pos = (lab != 0);
        neg_ws[row] = pos ? 0.0f : ce;
        if (pos) atomicAdd(&g_accum[1], ce);
    }
}

// ---------------- stage C: hard negative mining (bitonic top-k sum) ----------------
__global__ __launch_bounds__(256)
void hardneg_kernel(const float* __restrict__ neg_ws,
                    const int*   __restrict__ n_pos_img,
                    float* __restrict__ g_accum) {
    extern __shared__ float s[];                     // 16384 floats (dynamic LDS)
    __shared__ float red[256];
    const int SZ = 16384;
    const int b = blockIdx.x, tid = threadIdx.x;

    for (int i = tid; i < SZ; i += 256)
        s[i] = (i < P_) ? neg_ws[(size_t)b * P_ + i] : -1.0f;  // CE >= 0, pad sorts last
    __syncthreads();

    for (int ksz = 2; ksz <= SZ; ksz <<= 1) {
        for (int j = ksz >> 1; j > 0; j >>= 1) {
            for (int i = tid; i < SZ; i += 256) {
                int ixj = i ^ j;
                if (ixj > i) {
                    bool descend = ((i & ksz) == 0);
                    float a = s[i], c = s[ixj];
                    if (descend ? (a < c) : (a > c)) { s[i] = c; s[ixj] = a; }
                }
            }
            __syncthreads();
        }
    }

    int K = n_pos_img[b] * 3;
    if (K > P_) K = P_;
    float loc = 0.f;
    for (int i = tid; i < K; i += 256) loc += s[i];
    red[tid] = loc; __syncthreads();
    for (int off = 128; off > 0; off >>= 1) {
        if (tid < off) red[tid] += red[tid + off];
        __syncthreads();
    }
    if (tid == 0) atomicAdd(&g_accum[2], red[0]);
}

// ---------------- finalize ----------------
__global__ void finalize_kernel(const float* __restrict__ g_accum,
                                const int*   __restrict__ g_npos,
                                float* __restrict__ out) {
    if (threadIdx.x == 0 && blockIdx.x == 0) {
        int np_i = *g_npos; if (np_i < 1) np_i = 1;
        float np = (float)np_i;
        float cls  = (g_accum[1] + g_accum[2]) / np;
        float bbox = g_accum[0] / (np * 4.0f);
        out[0] = cls + bbox;
    }
}

extern "C" void kernel_launch(void* const* d_in, const int* in_sizes, int n_in,
                              void* d_out, int out_size, void* d_ws, size_t ws_size,
                              hipStream_t stream) {
    const float* pred_boxes  = (const float*)d_in[0];  // [128,8732,4]
    const float* pred_scores = (const float*)d_in[1];  // [128,8732,81]
    const float* tgt_boxes   = (const float*)d_in[2];  // [128,16,4]
    const int*   tgt_labels  = (const int*)  d_in[3];  // [128,16]
    const float* def_boxes   = (const float*)d_in[4];  // [8732,4]
    float* out = (float*)d_out;

    // workspace layout
    char* ws = (char*)d_ws;
    float* g_accum   = (float*)ws;                             // 4 floats
    int*   g_npos    = (int*)(ws + 16);                        // 1 int
    int*   n_pos_img = (int*)(ws + 64);                        // 128 ints
    int*   labels_ws = (int*)(ws + 1024);                      // B*P ints
    float* neg_ws    = (float*)(ws + 1024 + (size_t)B_*P_*4);  // B*P floats

    init_kernel<<<1, 64, 0, stream>>>(g_accum, g_npos);
    match_kernel<<<B_, 256, 0, stream>>>(pred_boxes, tgt_boxes, tgt_labels, def_boxes,
                                         labels_ws, n_pos_img, g_accum, g_npos);
    // (B*P)/16 tiles, 8 waves per block -> 8732 blocks exactly
    ce_kernel<<<(B_*P_)/(16*8), 256, 0, stream>>>(pred_scores, labels_ws, neg_ws, g_accum);
    hardneg_kernel<<<B_, 256, 16384*sizeof(float), stream>>>(neg_ws, n_pos_img, g_accum);
    finalize_kernel<<<1, 32, 0, stream>>>(g_accum, g_npos, out);
}